// latentODE_24180665876492
// MI455X (gfx1250) — compile-verified
//
#include <hip/hip_runtime.h>
#include <math.h>

// Problem constants (from reference)
#define T_TOT  200
#define BATCH  1024
#define DIM    128
#define TOBS   140
#define LDIM   20
#define LATD   32
#define GRUU   200
#define UNITS  100
#define GIN    296      // 2L + 2D
#define NWAVES 8
#define NTHREADS 256

typedef __attribute__((ext_vector_type(2))) float v2f;
typedef __attribute__((ext_vector_type(8))) float v8f;

enum { ACT_NONE = 0, ACT_TANH = 1, ACT_SIG = 2 };

// ---- gfx1250 hardware transcendentals (V_TANH_F32 / V_EXP_F32 / V_RCP_F32) ----
__device__ __forceinline__ float fast_tanh(float x)
{
#if __has_builtin(__builtin_amdgcn_tanhf)
    return __builtin_amdgcn_tanhf(x);       // single V_TANH_F32, no branches
#else
    return tanhf(x);
#endif
}

__device__ __forceinline__ float fast_sigmoid(float x)
{
    // 1/(1 + 2^(-x*log2(e))) : V_EXP_F32 is base-2 on AMD
    float e = __builtin_amdgcn_exp2f(-1.44269504088896340736f * x);
#if __has_builtin(__builtin_amdgcn_rcpf)
    return __builtin_amdgcn_rcpf(1.0f + e); // V_RCP_F32
#else
    return 1.0f / (1.0f + e);
#endif
}

// D(16xN) = A(16xK, LDS, row-major lda) @ B(KxN, global row-major) + bias, with
// optional activation. M=16 handled by one WMMA tile; N tiled by 16 across the
// 8 waves of the block. All K here are multiples of 4 (V_WMMA_F32_16X16X4_F32).
// Branches are wave-uniform so EXEC is all-ones at every WMMA (ISA 7.12 req).
__device__ __forceinline__ void gemm16(const float* __restrict__ A, int lda, int K,
                                       const float* __restrict__ B,
                                       const float* __restrict__ bias, int N,
                                       float* __restrict__ Out, long ldo, int act)
{
    const int lane = threadIdx.x & 31;
    const int wave = threadIdx.x >> 5;
    const int m    = lane & 15;   // M-row for A frag / N-col offset for B frag
    const int kh   = lane >> 4;   // selects K pair (0 -> K0,K1 ; 1 -> K2,K3)

    for (int n0 = wave * 16; n0 < N; n0 += NWAVES * 16) {
        const int col  = n0 + m;
        const int colc = (col < N) ? col : (N - 1);   // clamp to avoid OOB loads
        v8f c;
        {
            const float bv = bias[colc];
            #pragma unroll
            for (int r = 0; r < 8; ++r) c[r] = bv;    // bias broadcast down rows
        }
        for (int k = 0; k < K; k += 4) {
            v2f a, b;
            // A 16x4 f32 layout: lanes 0-15: K={0,1}; lanes 16-31: K={2,3}
            a[0] = A[m * lda + k + 2 * kh];
            a[1] = A[m * lda + k + 2 * kh + 1];
            // B 4x16 f32 layout mirrored: per-lane column, K pair by kh
            const float* Bp = B + (size_t)(k + 2 * kh) * N + colc;
            b[0] = Bp[0];
            b[1] = Bp[N];
            c = __builtin_amdgcn_wmma_f32_16x16x4_f32(
                    false, a, false, b, (short)0, c, false, false);
        }
        if (col < N) {
            #pragma unroll
            for (int r = 0; r < 8; ++r) {
                float v = c[r];
                if (act == ACT_TANH)      v = fast_tanh(v);
                else if (act == ACT_SIG)  v = fast_sigmoid(v);
                Out[(long)(r + 8 * kh) * ldo + col] = v;
            }
        }
    }
}

__global__ __launch_bounds__(NTHREADS)
void latent_ode_kernel(const float* __restrict__ truth, const float* __restrict__ tarr,
                       const int*   __restrict__ obs_idx, const float* __restrict__ eps,
                       const float* __restrict__ enc_w1, const float* __restrict__ enc_b1,
                       const float* __restrict__ enc_w2, const float* __restrict__ enc_b2,
                       const float* __restrict__ gru_uw1, const float* __restrict__ gru_ub1,
                       const float* __restrict__ gru_uw2, const float* __restrict__ gru_ub2,
                       const float* __restrict__ gru_rw1, const float* __restrict__ gru_rb1,
                       const float* __restrict__ gru_rw2, const float* __restrict__ gru_rb2,
                       const float* __restrict__ gru_nw1, const float* __restrict__ gru_nb1,
                       const float* __restrict__ gru_nw2, const float* __restrict__ gru_nb2,
                       const float* __restrict__ z0_w1, const float* __restrict__ z0_b1,
                       const float* __restrict__ z0_w2, const float* __restrict__ z0_b2,
                       const float* __restrict__ dec_w1, const float* __restrict__ dec_b1,
                       const float* __restrict__ dec_w2, const float* __restrict__ dec_b2,
                       const float* __restrict__ dec_w3, const float* __restrict__ dec_b3,
                       const float* __restrict__ out_w, const float* __restrict__ out_b,
                       float* __restrict__ out)
{
    __shared__ float s_y  [16 * 40];     // [ym | ys]
    __shared__ float s_yc [16 * GIN];    // GRU A-matrix: [y or y*r | x | ones]
    __shared__ float s_hid[16 * GRUU];   // gate hidden (also z0 head output)
    __shared__ float s_u  [16 * 40];
    __shared__ float s_r  [16 * 40];
    __shared__ float s_ns [16 * 40];
    __shared__ float s_t1 [16 * UNITS];
    __shared__ float s_t2 [16 * UNITS];
    __shared__ float s_in [16 * 40];     // RK4 eval input (stride 40)
    __shared__ float s_k  [16 * 40];     // RK4 slope
    __shared__ float s_acc[16 * 40];     // RK4 accumulator
    __shared__ float s_z  [16 * LATD];   // latent state (stride 32)

    const int tid = threadIdx.x;
    const int b0  = blockIdx.x * 16;

    for (int i = tid; i < 16 * 40; i += NTHREADS) s_y[i] = 0.0f;
    __syncthreads();

    // ======================= encoder scan (reversed obs) =======================
    for (int step = 0; step < TOBS; ++step) {
        const int oi = obs_idx[TOBS - 1 - step];
        float dt = 0.0f;
        if (step > 0)
            dt = tarr[obs_idx[TOBS - 1 - step]] - tarr[obs_idx[TOBS - step]];

        // ---- RK4 on ym (cols 0..19 of s_y) with f = enc MLP ----
        for (int i = tid; i < 16 * LDIM; i += NTHREADS) {
            int rr = i / LDIM, cc = i % LDIM;
            s_in[rr * 40 + cc] = s_y[rr * 40 + cc];
        }
        __syncthreads();
        // k1
        gemm16(s_in, 40, LDIM, enc_w1, enc_b1, UNITS, s_t1, UNITS, ACT_TANH); __syncthreads();
        gemm16(s_t1, UNITS, UNITS, enc_w2, enc_b2, LDIM, s_k, 40, ACT_NONE); __syncthreads();
        for (int i = tid; i < 16 * LDIM; i += NTHREADS) {
            int rr = i / LDIM, cc = i % LDIM, ix = rr * 40 + cc;
            s_acc[ix] = s_k[ix];
            s_in[ix]  = s_y[ix] + 0.5f * dt * s_k[ix];
        }
        __syncthreads();
        // k2
        gemm16(s_in, 40, LDIM, enc_w1, enc_b1, UNITS, s_t1, UNITS, ACT_TANH); __syncthreads();
        gemm16(s_t1, UNITS, UNITS, enc_w2, enc_b2, LDIM, s_k, 40, ACT_NONE); __syncthreads();
        for (int i = tid; i < 16 * LDIM; i += NTHREADS) {
            int rr = i / LDIM, cc = i % LDIM, ix = rr * 40 + cc;
            s_acc[ix] += 2.0f * s_k[ix];
            s_in[ix]   = s_y[ix] + 0.5f * dt * s_k[ix];
        }
        __syncthreads();
        // k3
        gemm16(s_in, 40, LDIM, enc_w1, enc_b1, UNITS, s_t1, UNITS, ACT_TANH); __syncthreads();
        gemm16(s_t1, UNITS, UNITS, enc_w2, enc_b2, LDIM, s_k, 40, ACT_NONE); __syncthreads();
        for (int i = tid; i < 16 * LDIM; i += NTHREADS) {
            int rr = i / LDIM, cc = i % LDIM, ix = rr * 40 + cc;
            s_acc[ix] += 2.0f * s_k[ix];
            s_in[ix]   = s_y[ix] + dt * s_k[ix];
        }
        __syncthreads();
        // k4 + combine
        gemm16(s_in, 40, LDIM, enc_w1, enc_b1, UNITS, s_t1, UNITS, ACT_TANH); __syncthreads();
        gemm16(s_t1, UNITS, UNITS, enc_w2, enc_b2, LDIM, s_k, 40, ACT_NONE); __syncthreads();
        for (int i = tid; i < 16 * LDIM; i += NTHREADS) {
            int rr = i / LDIM, cc = i % LDIM, ix = rr * 40 + cc;
            s_y[ix] += (dt / 6.0f) * (s_acc[ix] + s_k[ix]);
        }
        __syncthreads();

        // ---- build yc = [ym, ys, x, 1] ----
        for (int i = tid; i < 16 * GIN; i += NTHREADS) {
            int rr = i / GIN, cc = i % GIN;
            float v;
            if (cc < 40)            v = s_y[rr * 40 + cc];
            else if (cc < 40 + DIM) v = truth[(size_t)oi * BATCH * DIM
                                             + (size_t)(b0 + rr) * DIM + (cc - 40)];
            else                    v = 1.0f;
            s_yc[i] = v;
        }
        __syncthreads();

        // ---- u gate ----
        gemm16(s_yc, GIN, GIN, gru_uw1, gru_ub1, GRUU, s_hid, GRUU, ACT_TANH); __syncthreads();
        gemm16(s_hid, GRUU, GRUU, gru_uw2, gru_ub2, 40, s_u, 40, ACT_SIG); __syncthreads();
        // ---- r gate ----
        gemm16(s_yc, GIN, GIN, gru_rw1, gru_rb1, GRUU, s_hid, GRUU, ACT_TANH); __syncthreads();
        gemm16(s_hid, GRUU, GRUU, gru_rw2, gru_rb2, 40, s_r, 40, ACT_SIG); __syncthreads();
        // ---- n gate: overwrite y-part of yc with y*r ----
        for (int i = tid; i < 16 * 40; i += NTHREADS) {
            int rr = i / 40, cc = i % 40;
            s_yc[rr * GIN + cc] = s_y[i] * s_r[i];
        }
        __syncthreads();
        gemm16(s_yc, GIN, GIN, gru_nw1, gru_nb1, GRUU, s_hid, GRUU, ACT_TANH); __syncthreads();
        gemm16(s_hid, GRUU, GRUU, gru_nw2, gru_nb2, 40, s_ns, 40, ACT_NONE); __syncthreads();
        // ---- gated update: new_y = (1-u)*[nm,|nsd|] + u*y ----
        for (int i = tid; i < 16 * 40; i += NTHREADS) {
            int cc = i % 40;
            float nv = (cc < LDIM) ? s_ns[i] : fabsf(s_ns[i]);
            float uv = s_u[i];
            s_y[i] = (1.0f - uv) * nv + uv * s_y[i];
        }
        __syncthreads();
    }

    // ======================= z0 head + reparameterize =======================
    for (int i = tid; i < 16 * 40; i += NTHREADS) s_in[i] = s_y[i];
    __syncthreads();
    gemm16(s_in, 40, 40, z0_w1, z0_b1, UNITS, s_t1, UNITS, ACT_TANH); __syncthreads();
    gemm16(s_t1, UNITS, UNITS, z0_w2, z0_b2, 2 * LATD, s_hid, 2 * LATD, ACT_NONE); __syncthreads();
    for (int i = tid; i < 16 * LATD; i += NTHREADS) {
        int rr = i / LATD, cc = i % LATD;
        float mean = s_hid[rr * 2 * LATD + cc];
        float sd   = fabsf(s_hid[rr * 2 * LATD + cc + LATD]);
        s_z[i] = mean + eps[(size_t)(b0 + rr) * LATD + cc] * sd;
    }
    __syncthreads();

    // output at t=0  (out[0][b][t][d], row stride T*D)
    gemm16(s_z, LATD, LATD, out_w, out_b, DIM,
           out + (size_t)b0 * T_TOT * DIM, (long)T_TOT * DIM, ACT_NONE);
    __syncthreads();

    // ======================= decoder scan =======================
    for (int s = 0; s < T_TOT - 1; ++s) {
        const float dt = tarr[s + 1] - tarr[s];

        for (int i = tid; i < 16 * LATD; i += NTHREADS) {
            int rr = i / LATD, cc = i % LATD;
            s_in[rr * 40 + cc] = s_z[i];
        }
        __syncthreads();
        // k1
        gemm16(s_in, 40, LATD, dec_w1, dec_b1, UNITS, s_t1, UNITS, ACT_TANH); __syncthreads();
        gemm16(s_t1, UNITS, UNITS, dec_w2, dec_b2, UNITS, s_t2, UNITS, ACT_TANH); __syncthreads();
        gemm16(s_t2, UNITS, UNITS, dec_w3, dec_b3, LATD, s_k, 40, ACT_NONE); __syncthreads();
        for (int i = tid; i < 16 * LATD; i += NTHREADS) {
            int rr = i / LATD, cc = i % LATD, ix = rr * 40 + cc;
            s_acc[ix] = s_k[ix];
            s_in[ix]  = s_z[i] + 0.5f * dt * s_k[ix];
        }
        __syncthreads();
        // k2
        gemm16(s_in, 40, LATD, dec_w1, dec_b1, UNITS, s_t1, UNITS, ACT_TANH); __syncthreads();
        gemm16(s_t1, UNITS, UNITS, dec_w2, dec_b2, UNITS, s_t2, UNITS, ACT_TANH); __syncthreads();
        gemm16(s_t2, UNITS, UNITS, dec_w3, dec_b3, LATD, s_k, 40, ACT_NONE); __syncthreads();
        for (int i = tid; i < 16 * LATD; i += NTHREADS) {
            int rr = i / LATD, cc = i % LATD, ix = rr * 40 + cc;
            s_acc[ix] += 2.0f * s_k[ix];
            s_in[ix]   = s_z[i] + 0.5f * dt * s_k[ix];
        }
        __syncthreads();
        // k3
        gemm16(s_in, 40, LATD, dec_w1, dec_b1, UNITS, s_t1, UNITS, ACT_TANH); __syncthreads();
        gemm16(s_t1, UNITS, UNITS, dec_w2, dec_b2, UNITS, s_t2, UNITS, ACT_TANH); __syncthreads();
        gemm16(s_t2, UNITS, UNITS, dec_w3, dec_b3, LATD, s_k, 40, ACT_NONE); __syncthreads();
        for (int i = tid; i < 16 * LATD; i += NTHREADS) {
            int rr = i / LATD, cc = i % LATD, ix = rr * 40 + cc;
            s_acc[ix] += 2.0f * s_k[ix];
            s_in[ix]   = s_z[i] + dt * s_k[ix];
        }
        __syncthreads();
        // k4 + combine
        gemm16(s_in, 40, LATD, dec_w1, dec_b1, UNITS, s_t1, UNITS, ACT_TANH); __syncthreads();
        gemm16(s_t1, UNITS, UNITS, dec_w2, dec_b2, UNITS, s_t2, UNITS, ACT_TANH); __syncthreads();
        gemm16(s_t2, UNITS, UNITS, dec_w3, dec_b3, LATD, s_k, 40, ACT_NONE); __syncthreads();
        for (int i = tid; i < 16 * LATD; i += NTHREADS) {
            int rr = i / LATD, cc = i % LATD, ix = rr * 40 + cc;
            s_z[i] += (dt / 6.0f) * (s_acc[ix] + s_k[ix]);
        }
        __syncthreads();

        // output at t = s+1
        gemm16(s_z, LATD, LATD, out_w, out_b, DIM,
               out + (size_t)b0 * T_TOT * DIM + (size_t)(s + 1) * DIM,
               (long)T_TOT * DIM, ACT_NONE);
        __syncthreads();
    }
}

extern "C" void kernel_launch(void* const* d_in, const int* in_sizes, int n_in,
                              void* d_out, int out_size, void* d_ws, size_t ws_size,
                              hipStream_t stream)
{
    (void)in_sizes; (void)n_in; (void)out_size; (void)d_ws; (void)ws_size;
    const float* truth   = (const float*)d_in[0];
    const float* tarr    = (const float*)d_in[1];
    const int*   obs_idx = (const int*)  d_in[2];
    const float* eps     = (const float*)d_in[3];
    const float* enc_w1  = (const float*)d_in[4];
    const float* enc_b1  = (const float*)d_in[5];
    const float* enc_w2  = (const float*)d_in[6];
    const float* enc_b2  = (const float*)d_in[7];
    const float* gru_uw1 = (const float*)d_in[8];
    const float* gru_ub1 = (const float*)d_in[9];
    const float* gru_uw2 = (const float*)d_in[10];
    const float* gru_ub2 = (const float*)d_in[11];
    const float* gru_rw1 = (const float*)d_in[12];
    const float* gru_rb1 = (const float*)d_in[13];
    const float* gru_rw2 = (const float*)d_in[14];
    const float* gru_rb2 = (const float*)d_in[15];
    const float* gru_nw1 = (const float*)d_in[16];
    const float* gru_nb1 = (const float*)d_in[17];
    const float* gru_nw2 = (const float*)d_in[18];
    const float* gru_nb2 = (const float*)d_in[19];
    const float* z0_w1   = (const float*)d_in[20];
    const float* z0_b1   = (const float*)d_in[21];
    const float* z0_w2   = (const float*)d_in[22];
    const float* z0_b2   = (const float*)d_in[23];
    const float* dec_w1  = (const float*)d_in[24];
    const float* dec_b1  = (const float*)d_in[25];
    const float* dec_w2  = (const float*)d_in[26];
    const float* dec_b2  = (const float*)d_in[27];
    const float* dec_w3  = (const float*)d_in[28];
    const float* dec_b3  = (const float*)d_in[29];
    const float* out_w   = (const float*)d_in[30];
    const float* out_b   = (const float*)d_in[31];

    dim3 grid(BATCH / 16);   // 64 blocks, 16 batch rows each
    dim3 block(NTHREADS);    // 8 waves

    hipLaunchKernelGGL(latent_ode_kernel, grid, block, 0, stream,
                       truth, tarr, obs_idx, eps,
                       enc_w1, enc_b1, enc_w2, enc_b2,
                       gru_uw1, gru_ub1, gru_uw2, gru_ub2,
                       gru_rw1, gru_rb1, gru_rw2, gru_rb2,
                       gru_nw1, gru_nb1, gru_nw2, gru_nb2,
                       z0_w1, z0_b1, z0_w2, z0_b2,
                       dec_w1, dec_b1, dec_w2, dec_b2, dec_w3, dec_b3,
                       out_w, out_b,
                       (float*)d_out);
}